// IGMC_61916248539607
// MI455X (gfx1250) — compile-verified
//
#include <hip/hip_runtime.h>
#include <math.h>

// ---------------- static problem sizes (match reference) ----------------
#define N_NODES 100000
#define B_GR    1000
#define NPG     100
#define E_TOT   1000000
#define EPG     1000
#define R_REL   5
#define F_INDIM 4
#define DLAT    32
#define L_LAY   4
#define NBAS    2
#define HIDD    128
#define NCLS    5

typedef __attribute__((ext_vector_type(2))) float v2f;
typedef __attribute__((ext_vector_type(8))) float v8f;

// ---------------- LDS layout (floats) ----------------
// edges(int)  : 1024
// H  [112][32]: 3584   (layer input, rows 100..111 zero padding for WMMA)
// HN [112][32]: 3584   (layer pre-norm output)
// ACC[112][32]: 3584   (per-relation segment sums -> means)
// CNT[112]    : 112
// RED[256]    : 256    (reductions; reused by MLP partials)
// MEANV[32], STDV[32], BIASV[32], GNW[32], GNB[32], GNMS[32] : 192
// SU[128], SV[128], H1[128], Z2[8] : 392
#define LDS_FLOATS (1024 + 3584 + 3584 + 3584 + 112 + 256 + 192 + 392)

struct IgmcParams {
    const float* x;
    const int*   ei;   // edge_index [2,E]
    const int*   et;   // edge_type  [E]
    const float* Wg;   // precomputed W[l][r][32][32] in d_ws
    const float* root[L_LAY];
    const float* bias[L_LAY];
    const float* gnw[L_LAY];
    const float* gnb[L_LAY];
    const float* gnms[L_LAY];
    const float* lin1w;
    const float* lin1b;
    const float* lin2w;
    const float* lin2b;
    float*       out;
};

// W[r] = comp[r,0]*basis[0] + comp[r,1]*basis[1]   (per layer)
__global__ void igmc_wcomp(const float* __restrict__ basis,
                           const float* __restrict__ comp,
                           float* __restrict__ W, int din) {
    int idx = blockIdx.x * blockDim.x + threadIdx.x;
    int tot = R_REL * din * DLAT;
    if (idx >= tot) return;
    int r   = idx / (din * DLAT);
    int rem = idx % (din * DLAT);          // rem = i*32 + o
    float w = comp[r * NBAS + 0] * basis[rem]
            + comp[r * NBAS + 1] * basis[din * DLAT + rem];
    W[r * DLAT * DLAT + rem] = w;          // fixed stride 32 rows per relation
}

__global__ __launch_bounds__(256)
void igmc_fused(IgmcParams p) {
    extern __shared__ float sm[];
    int*   edges = (int*)sm;               // [1024]
    float* H     = sm + 1024;              // [112*32]
    float* HN    = H + 3584;
    float* ACC   = HN + 3584;
    float* CNT   = ACC + 3584;             // [112]
    float* RED   = CNT + 112;              // [256]
    float* MEANV = RED + 256;              // [32]
    float* STDV  = MEANV + 32;             // [32]
    float* BIASV = STDV + 32;              // [32]
    float* GNWv  = BIASV + 32;             // [32]
    float* GNBv  = GNWv + 32;              // [32]
    float* GNMSv = GNBv + 32;              // [32]
    float* SU    = GNMSv + 32;             // [128] user-node states (node 0)
    float* SV    = SU + 128;               // [128] item-node states (node 1)
    float* H1    = SV + 128;               // [128]
    float* Z2    = H1 + 128;               // [8]

    const int tid  = threadIdx.x;
    const int g    = blockIdx.x;
    const int wave = tid >> 5;
    const int lane = tid & 31;
    const int ml   = lane & 15;            // WMMA M / N lane index
    const int koff = (lane >> 4) << 1;     // K offset 0 or 2 per f32 fragment layout
    const int mhi  = (lane >> 4) << 3;     // +8 rows for upper lane half (C layout)

    // ---- stage packed edges (src|dst|rel), local node ids ----
    for (int e = tid; e < EPG; e += 256) {
        int s = p.ei[g * EPG + e]         - g * NPG;
        int d = p.ei[E_TOT + g * EPG + e] - g * NPG;
        int t = p.et[g * EPG + e];
        edges[e] = s | (d << 8) | (t << 16);
    }
    // ---- init H: zero (incl. padding rows), then scatter x one-hots ----
    for (int i = tid; i < 112 * 32; i += 256) H[i] = 0.0f;
    __syncthreads();
    for (int i = tid; i < NPG * F_INDIM; i += 256) {
        int n = i >> 2, c = i & 3;
        H[n * 32 + c] = p.x[(g * NPG + n) * F_INDIM + c];
    }
    __syncthreads();

    for (int l = 0; l < L_LAY; ++l) {
        const int din = (l == 0) ? F_INDIM : DLAT;
        const int k4n = din >> 2;
        const float* Wl = p.Wg + l * (R_REL * DLAT * DLAT);

        if (tid < 32) {
            BIASV[tid] = p.bias[l][tid];
            GNWv[tid]  = p.gnw[l][tid];
            GNBv[tid]  = p.gnb[l][tid];
            GNMSv[tid] = p.gnms[l][tid];
        }

        // per-wave WMMA accumulators: up to 2 (mt,nt) jobs per wave
        v8f c0 = {}; v8f c1 = {};

        // -------- relation loop: segment-mean then WMMA accumulate --------
        for (int r = 0; r < R_REL; ++r) {
            for (int i = tid; i < 112 * 32; i += 256) ACC[i] = 0.0f;
            for (int i = tid; i < 112;      i += 256) CNT[i] = 0.0f;
            __syncthreads();

            for (int e = tid; e < EPG; e += 256) {
                int pk = edges[e];
                if (((pk >> 16) & 7) == r) {
                    int s = pk & 255, d = (pk >> 8) & 255;
                    for (int i = 0; i < din; ++i)
                        atomicAdd(&ACC[d * 32 + i], H[s * 32 + i]);
                    atomicAdd(&CNT[d], 1.0f);
                }
            }
            __syncthreads();

            for (int i = tid; i < NPG * din; i += 256) {
                int n = i / din, k = i % din;
                ACC[n * 32 + k] /= fmaxf(CNT[n], 1.0f);
            }
            __syncthreads();

            // WMMA: C += mean_r[16x4] x W[r][4x16], K-sliced
            const float* Wr = Wl + r * (DLAT * DLAT);
            for (int j = 0; j < 2; ++j) {
                int job = wave + 8 * j;                 // wave-uniform
                if (job < 14) {
                    int mt = job % 7, nt = job / 7;
                    v8f acc = (j == 0) ? c0 : c1;
                    for (int k4 = 0; k4 < k4n; ++k4) {
                        int kb = k4 * 4 + koff;
                        v2f a, b;
                        a.x = ACC[(mt * 16 + ml) * 32 + kb];
                        a.y = ACC[(mt * 16 + ml) * 32 + kb + 1];
                        b.x = Wr[kb * DLAT       + nt * 16 + ml];
                        b.y = Wr[(kb + 1) * DLAT + nt * 16 + ml];
                        acc = __builtin_amdgcn_wmma_f32_16x16x4_f32(
                                  false, a, false, b, (short)0, acc, false, false);
                    }
                    if (j == 0) c0 = acc; else c1 = acc;
                }
            }
            __syncthreads();
        }

        // -------- root term + bias, store to HN --------
        {
            const float* rw = p.root[l];
            for (int j = 0; j < 2; ++j) {
                int job = wave + 8 * j;
                if (job < 14) {
                    int mt = job % 7, nt = job / 7;
                    v8f acc = (j == 0) ? c0 : c1;
                    for (int k4 = 0; k4 < k4n; ++k4) {
                        int kb = k4 * 4 + koff;
                        v2f a, b;
                        a.x = H[(mt * 16 + ml) * 32 + kb];
                        a.y = H[(mt * 16 + ml) * 32 + kb + 1];
                        b.x = rw[kb * DLAT       + nt * 16 + ml];
                        b.y = rw[(kb + 1) * DLAT + nt * 16 + ml];
                        acc = __builtin_amdgcn_wmma_f32_16x16x4_f32(
                                  false, a, false, b, (short)0, acc, false, false);
                    }
                    float bs = BIASV[nt * 16 + ml];
                    #pragma unroll
                    for (int v = 0; v < 8; ++v) {
                        int n = mt * 16 + v + mhi;
                        if (n < NPG) HN[n * 32 + nt * 16 + ml] = acc[v] + bs;
                    }
                }
            }
        }
        __syncthreads();

        // -------- GraphNorm (+tanh) over 100 nodes x 32 channels --------
        {
            int cch = tid & 31, ph = tid >> 5;
            float s = 0.0f;
            for (int n = ph; n < NPG; n += 8) s += HN[n * 32 + cch];
            RED[ph * 32 + cch] = s;
            __syncthreads();
            if (tid < 32) {
                float m = 0.0f;
                for (int q = 0; q < 8; ++q) m += RED[q * 32 + tid];
                MEANV[tid] = m * (1.0f / NPG);
            }
            __syncthreads();
            float mm = MEANV[cch] * GNMSv[cch];
            s = 0.0f;
            for (int n = ph; n < NPG; n += 8) {
                float sb = HN[n * 32 + cch] - mm;
                s += sb * sb;
            }
            RED[ph * 32 + cch] = s;
            __syncthreads();
            if (tid < 32) {
                float v = 0.0f;
                for (int q = 0; q < 8; ++q) v += RED[q * 32 + tid];
                STDV[tid] = sqrtf(v * (1.0f / NPG) + 1e-6f);
            }
            __syncthreads();
            for (int i = tid; i < NPG * 32; i += 256) {
                int cc = i & 31;
                float sb = HN[i] - MEANV[cc] * GNMSv[cc];
                H[i] = tanhf(GNWv[cc] * sb / STDV[cc] + GNBv[cc]);
            }
            __syncthreads();
            if (tid < 32) {                 // save user/item node states
                SU[l * 32 + tid] = H[tid];           // local node 0
                SV[l * 32 + tid] = H[32 + tid];      // local node 1
            }
            __syncthreads();
        }
    }

    // -------- readout MLP: z[256]=[SU,SV] -> 128 relu -> 5 -> log_softmax --------
    {
        int j = tid & 127, half = tid >> 7;
        float s = 0.0f;
        for (int i = 0; i < 128; ++i) {
            float zv = (half == 0) ? SU[i] : SV[i];
            s += zv * p.lin1w[(half * 128 + i) * HIDD + j];
        }
        RED[half * 128 + j] = s;
        __syncthreads();
        if (tid < HIDD) {
            float v = p.lin1b[tid] + RED[tid] + RED[128 + tid];
            H1[tid] = fmaxf(v, 0.0f);
        }
        __syncthreads();
        if (tid < NCLS) {
            float s2 = p.lin2b[tid];
            for (int jj = 0; jj < HIDD; ++jj)
                s2 += H1[jj] * p.lin2w[jj * NCLS + tid];
            Z2[tid] = s2;
        }
        __syncthreads();
        if (tid == 0) {
            float m = Z2[0];
            for (int c = 1; c < NCLS; ++c) m = fmaxf(m, Z2[c]);
            float se = 0.0f;
            for (int c = 0; c < NCLS; ++c) se += expf(Z2[c] - m);
            float ls = m + logf(se);
            for (int c = 0; c < NCLS; ++c) p.out[g * NCLS + c] = Z2[c] - ls;
        }
    }
}

extern "C" void kernel_launch(void* const* d_in, const int* in_sizes, int n_in,
                              void* d_out, int out_size, void* d_ws, size_t ws_size,
                              hipStream_t stream) {
    (void)in_sizes; (void)n_in; (void)out_size; (void)ws_size;
    // input order (setup_inputs insertion order):
    // 0:x 1:edge_index 2:edge_type 3:batch(unused)
    // then per layer l: 4+7l..: basis, comp, root, bias, gn_w, gn_b, gn_ms
    // then 32:lin1_w 33:lin1_b 34:lin2_w 35:lin2_b
    IgmcParams p;
    p.x  = (const float*)d_in[0];
    p.ei = (const int*)  d_in[1];
    p.et = (const int*)  d_in[2];
    float* Wg = (float*)d_ws;              // 4*5*32*32 floats = 80 KB scratch
    p.Wg = Wg;
    for (int l = 0; l < L_LAY; ++l) {
        void* const* Lp = d_in + 4 + 7 * l;
        const float* basis = (const float*)Lp[0];
        const float* comp  = (const float*)Lp[1];
        p.root[l] = (const float*)Lp[2];
        p.bias[l] = (const float*)Lp[3];
        p.gnw[l]  = (const float*)Lp[4];
        p.gnb[l]  = (const float*)Lp[5];
        p.gnms[l] = (const float*)Lp[6];
        int din = (l == 0) ? F_INDIM : DLAT;
        int tot = R_REL * din * DLAT;
        igmc_wcomp<<<(tot + 255) / 256, 256, 0, stream>>>(
            basis, comp, Wg + l * (R_REL * DLAT * DLAT), din);
    }
    p.lin1w = (const float*)d_in[32];
    p.lin1b = (const float*)d_in[33];
    p.lin2w = (const float*)d_in[34];
    p.lin2b = (const float*)d_in[35];
    p.out   = (float*)d_out;

    size_t shmem = (size_t)LDS_FLOATS * sizeof(float);   // ~50 KB / block
    igmc_fused<<<B_GR, 256, shmem, stream>>>(p);
}